// SAGE_893353198160
// MI455X (gfx1250) — compile-verified
//
#include <hip/hip_runtime.h>

typedef float v2f __attribute__((ext_vector_type(2)));
typedef float v8f __attribute__((ext_vector_type(8)));

#define NNODES 100000
#define NEDGES 1000000
#define DIM    64

// ---------------------------------------------------------------- utilities
__global__ void zero_f32(float* __restrict__ p, int n) {
    int i = blockIdx.x * blockDim.x + threadIdx.x;
    if (i < n) p[i] = 0.0f;
}

// deg[v] = number of incoming edges (as float, for the mean divisor)
__global__ void degree_kernel(const int* __restrict__ dst, float* __restrict__ deg, int E) {
    int e = blockIdx.x * blockDim.x + threadIdx.x;
    if (e < E) atomicAdd(&deg[dst[e]], 1.0f);
}

// agg[dst] += h[src]  — one thread per (edge, 4-float chunk): 128-bit coalesced
// gather + 4 no-return fp32 atomics. h and agg are L2-resident (25.6MB each).
__global__ void scatter_kernel(const float* __restrict__ h,
                               const int* __restrict__ src,
                               const int* __restrict__ dst,
                               float* __restrict__ agg, int E) {
    long t = (long)blockIdx.x * blockDim.x + threadIdx.x;
    if (t >= (long)E * 16) return;
    int e = (int)(t >> 4);
    int c = ((int)t & 15) * 4;
    int s = src[e];
    int d = dst[e];
    const float4 v = *(const float4*)(h + (long)s * DIM + c);
    float* a = agg + (long)d * DIM + c;
    atomicAdd(a + 0, v.x);
    atomicAdd(a + 1, v.y);
    atomicAdd(a + 2, v.z);
    atomicAdd(a + 3, v.w);
}

// One SAGEConv layer, eval mode:
//   hout = [relu]( (agg * inv_deg) @ Wl^T + bl + hin @ Wr^T )
// One wave32 per 16-row tile. fp32 WMMA 16x16x4, K=64 in 16 steps, two weight
// matrices -> 32 WMMA per 16x16 output tile, 4 col tiles -> 128 WMMA per wave.
//
// A layout (32-bit A 16x4, ISA 7.12.2): lane L holds row M=L%16,
//   VGPR0 = K = 2*(L/16), VGPR1 = K+1  -> one aligned float2 load per k-step.
// B layout mirrors with N = L%16; B[k][n] = W[n*64+k] -> contiguous float2.
// C/D layout: VGPR v holds row M = v + 8*(L/16), col N = L%16.
__global__ void __launch_bounds__(128)
sage_layer_kernel(const float* __restrict__ agg, const float* __restrict__ hin,
                  const float* __restrict__ deg,
                  const float* __restrict__ Wl, const float* __restrict__ bl,
                  const float* __restrict__ Wr,
                  float* __restrict__ hout, int relu, int n) {
    const int lane    = threadIdx.x & 31;
    const int wave    = threadIdx.x >> 5;
    const int rowBase = (blockIdx.x * 4 + wave) * 16;
    if (rowBase >= n) return;            // wave-uniform: EXEC stays all-1s for WMMA

    const int mrow = rowBase + (lane & 15);
    const int koff = (lane >> 4) * 2;    // K offset within 4-wide step per lane half
    const float invd = 1.0f / fmaxf(deg[mrow], 1.0f);

    // Hoist both A operands (mean-aggregated neighbors and root features).
    v2f aagg[16], ah[16];
    const float* ar = agg + (long)mrow * DIM;
    const float* hr = hin + (long)mrow * DIM;
#pragma unroll
    for (int kb = 0; kb < 16; ++kb) {
        const int k = kb * 4 + koff;
        v2f t = *(const v2f*)(ar + k);
        t.x *= invd;
        t.y *= invd;
        aagg[kb] = t;
        ah[kb]   = *(const v2f*)(hr + k);
    }

#pragma unroll
    for (int nt = 0; nt < 4; ++nt) {
        const int col = nt * 16 + (lane & 15);
        const float bias = bl[col];
        v8f acc;
#pragma unroll
        for (int v = 0; v < 8; ++v) acc[v] = bias;   // bias depends on col only

        const float* wl = Wl + (long)col * DIM;      // Wl^T[k][col] = Wl[col*64+k]
        const float* wr = Wr + (long)col * DIM;
#pragma unroll
        for (int kb = 0; kb < 16; ++kb) {
            v2f b = *(const v2f*)(wl + kb * 4 + koff);
            acc = __builtin_amdgcn_wmma_f32_16x16x4_f32(
                false, aagg[kb], false, b, (short)0, acc, false, false);
        }
#pragma unroll
        for (int kb = 0; kb < 16; ++kb) {
            v2f b = *(const v2f*)(wr + kb * 4 + koff);
            acc = __builtin_amdgcn_wmma_f32_16x16x4_f32(
                false, ah[kb], false, b, (short)0, acc, false, false);
        }

        if (relu) {
#pragma unroll
            for (int v = 0; v < 8; ++v) acc[v] = fmaxf(acc[v], 0.0f);
        }

        const int rstore = rowBase + 8 * (lane >> 4);
#pragma unroll
        for (int v = 0; v < 8; ++v)
            hout[(long)(rstore + v) * DIM + col] = acc[v];
    }
}

// out[m,:] = h[m,:] @ Wout^T + bout   (C=2: trivial VALU dot products)
__global__ void classify_kernel(const float* __restrict__ h,
                                const float* __restrict__ Wout,
                                const float* __restrict__ bout,
                                float* __restrict__ out, int n) {
    int m = blockIdx.x * blockDim.x + threadIdx.x;
    if (m >= n) return;
    float a0 = bout[0], a1 = bout[1];
    const float* hr = h + (long)m * DIM;
#pragma unroll 8
    for (int k = 0; k < DIM; ++k) {
        const float v = hr[k];
        a0 += v * Wout[k];
        a1 += v * Wout[DIM + k];
    }
    out[(long)m * 2 + 0] = a0;
    out[(long)m * 2 + 1] = a1;
}

// ---------------------------------------------------------------- launcher
extern "C" void kernel_launch(void* const* d_in, const int* in_sizes, int n_in,
                              void* d_out, int out_size, void* d_ws, size_t ws_size,
                              hipStream_t stream) {
    const float* x    = (const float*)d_in[0];   // [N,64]
    const int*   ei   = (const int*)d_in[1];     // [2,E]
    const float* Wl   = (const float*)d_in[2];   // [3,64,64]
    const float* bl   = (const float*)d_in[3];   // [3,64]
    const float* Wr   = (const float*)d_in[4];   // [3,64,64]
    const float* Wout = (const float*)d_in[5];   // [2,64]
    const float* bout = (const float*)d_in[6];   // [2]
    float* out = (float*)d_out;                  // [N,2]

    const int N = NNODES, E = NEDGES;
    const int* src = ei;       // edge_index[0]
    const int* dst = ei + E;   // edge_index[1]

    // Workspace layout (floats): deg[N] | agg[N*64] | h0[N*64] | h1[N*64]
    float* ws   = (float*)d_ws;
    float* deg  = ws;
    float* aggb = ws + 100032;                  // 16B-aligned region start
    float* h0   = aggb + (long)N * DIM;
    float* h1   = h0 + (long)N * DIM;

    zero_f32<<<(N + 255) / 256, 256, 0, stream>>>(deg, N);
    degree_kernel<<<(E + 255) / 256, 256, 0, stream>>>(dst, deg, E);

    const float* hin = x;
    float* houts[3] = {h0, h1, h0};
    for (int i = 0; i < 3; ++i) {
        const int aggElems = N * DIM;
        zero_f32<<<(aggElems + 255) / 256, 256, 0, stream>>>(aggb, aggElems);

        const long sthreads = (long)E * 16;
        scatter_kernel<<<(int)((sthreads + 255) / 256), 256, 0, stream>>>(
            hin, src, dst, aggb, E);

        const int waves  = N / 16;              // 6250 (N divisible by 16)
        const int blocks = (waves + 3) / 4;     // 4 waves / block
        sage_layer_kernel<<<blocks, 128, 0, stream>>>(
            aggb, hin, deg,
            Wl + (long)i * DIM * DIM, bl + (long)i * DIM,
            Wr + (long)i * DIM * DIM,
            houts[i], (i < 2) ? 1 : 0, N);

        hin = houts[i];
    }

    classify_kernel<<<(N + 255) / 256, 256, 0, stream>>>(hin, Wout, bout, out, N);
}